// DepthRenderer_76141180223956
// MI455X (gfx1250) — compile-verified
//
#include <hip/hip_runtime.h>
#include <cstdint>

// ---------------- problem constants (match reference) ----------------
#define HS     360
#define WS     640
#define NPIX   (HS * WS)
#define NSQ    8
#define NSAMP  10            // interior samples; total 12 with near/far
#define NEAR_T 0.0f
#define FAR_T  1.5f
#define SHARP  1000.0f
#define TAU    100.0f
#define EPS_F  1e-6f
#define LOG2E  1.44269504088896340736f

// LDS layout (floats)
#define L_POSE 0     // 8*16 = 128
#define L_PAR  128   // 8*5  = 40
#define L_T    168   // 10
#define L_O    178   // 3
#define L_TOT  181

// ---------------- fast HW transcendentals ----------------
__device__ __forceinline__ float fexp2(float x) { return __builtin_amdgcn_exp2f(x); }
__device__ __forceinline__ float flog2(float x) { return __builtin_amdgcn_logf(x); }   // log2
__device__ __forceinline__ float frsq (float x) { return __builtin_amdgcn_rsqf(x); }
__device__ __forceinline__ float frcp (float x) { return __builtin_amdgcn_rcpf(x); }
__device__ __forceinline__ float fsqrt_(float x){ return __builtin_amdgcn_sqrtf(x); }
// pow(x,e) for x > 0
__device__ __forceinline__ float fpow(float x, float e) { return fexp2(e * flog2(x)); }

// ---------------- CDNA5 async global->LDS staging ----------------
__device__ __forceinline__ void async_g2l_b32(const void* g, void* l) {
#if __has_builtin(__builtin_amdgcn_global_load_async_to_lds_b32)
    __builtin_amdgcn_global_load_async_to_lds_b32(
        (__attribute__((address_space(1))) int*)(uintptr_t)g,
        (__attribute__((address_space(3))) int*)(uint32_t)(uintptr_t)l,
        /*imm offset*/0, /*cpol*/0);
#else
    unsigned int       loff = (unsigned int)(uintptr_t)l;       // generic LDS ptr low 32 = LDS offset
    unsigned long long ga   = (unsigned long long)(uintptr_t)g;
    asm volatile("global_load_async_to_lds_b32 %0, %1, off"
                 :: "v"(loff), "v"(ga) : "memory");
#endif
}

__device__ __forceinline__ void wait_async0() {
#if __has_builtin(__builtin_amdgcn_s_wait_asynccnt)
    __builtin_amdgcn_s_wait_asynccnt(0);
#else
    asm volatile("s_wait_asynccnt 0" ::: "memory");
#endif
}

// ---------------- kernel ----------------
__global__ __launch_bounds__(256) void sq_depth_kernel(
    const float* __restrict__ poses,   // (8,4,4)
    const float* __restrict__ params,  // (8,5)  a0 a1 a2 e1 e2
    const float* __restrict__ rays_d,  // (HS,WS,3)
    const float* __restrict__ rays_o,  // (3,)
    const float* __restrict__ tvals,   // (10,)
    float* __restrict__ out)           // (HS,WS)
{
    __shared__ float smem[L_TOT + 3];

    const int tid = threadIdx.x;

    // ---- stage tiny constant tensors into LDS via async DMA (ASYNCcnt path) ----
    const float* src = nullptr;
    if      (tid < L_PAR)  src = poses  + tid;
    else if (tid < L_T)    src = params + (tid - L_PAR);
    else if (tid < L_O)    src = tvals  + (tid - L_T);
    else if (tid < L_TOT)  src = rays_o + (tid - L_O);
    if (src) async_g2l_b32(src, &smem[tid]);
    wait_async0();
    __syncthreads();

    const int pix = blockIdx.x * blockDim.x + tid;
    if (pix >= NPIX) return;

    const float d0 = rays_d[3 * pix + 0];
    const float d1 = rays_d[3 * pix + 1];
    const float d2 = rays_d[3 * pix + 2];
    const float o0 = smem[L_O + 0], o1 = smem[L_O + 1], o2 = smem[L_O + 2];

    float best = __builtin_inff();

#pragma unroll 1
    for (int n = 0; n < NSQ; ++n) {
        const float* P = &smem[L_POSE + n * 16];
        const float r00 = P[0],  r01 = P[1],  r02 = P[2],  p0 = P[3];
        const float r10 = P[4],  r11 = P[5],  r12 = P[6],  p1 = P[7];
        const float r20 = P[8],  r21 = P[9],  r22 = P[10], p2 = P[11];
        const float* Q = &smem[L_PAR + n * 5];
        const float a0 = Q[0], a1 = Q[1], a2 = Q[2], e1 = Q[3], e2 = Q[4];

        const float ia0 = 1.0f / a0, ia1 = 1.0f / a1, ia2 = 1.0f / a2;
        const float ie1 = 1.0f / e1, ie2 = 1.0f / e2;
        const float exy  = 2.0f * ie2;   // 2/e2
        const float ez   = 2.0f * ie1;   // 2/e1
        const float emid = e2 * ie1;     // e2/e1

        // td = (R^T d) / a, normalized
        float td0 = (r00 * d0 + r10 * d1 + r20 * d2) * ia0;
        float td1 = (r01 * d0 + r11 * d1 + r21 * d2) * ia1;
        float td2 = (r02 * d0 + r12 * d1 + r22 * d2) * ia2;
        const float rn = frsq(td0 * td0 + td1 * td1 + td2 * td2);
        td0 *= rn; td1 *= rn; td2 *= rn;

        // tc = R^T (o - p) / a
        const float m0 = o0 - p0, m1 = o1 - p1, m2 = o2 - p2;
        const float tc0 = (r00 * m0 + r10 * m1 + r20 * m2) * ia0;
        const float tc1 = (r01 * m0 + r11 * m1 + r21 * m2) * ia1;
        const float tc2 = (r02 * m0 + r12 * m1 + r22 * m2) * ia2;

        const float proj = fabsf(tc0 * td0 + tc1 * td1 + tc2 * td2);
        const float c0 = tc0 + proj * td0;
        const float c1 = tc1 + proj * td1;
        const float c2 = tc2 + proj * td2;
        const float dist2 = c0 * c0 + c1 * c1 + c2 * c2;
        const bool  mask  = dist2 < 3.0f;
        const float hcl   = fsqrt_(fmaxf(3.0f - dist2, 1e-12f));

        float cum = 0.0f, depth = 0.0f;
        float pv0 = 0.0f, pv1 = 0.0f, pv2 = 0.0f, prevvis = 0.0f;

#pragma unroll
        for (int s = 0; s < NSAMP + 2; ++s) {
            float pw0, pw1, pw2;
            if (s == 0) {
                pw0 = o0 + d0 * NEAR_T; pw1 = o1 + d1 * NEAR_T; pw2 = o2 + d2 * NEAR_T;
            } else if (s == NSAMP + 1) {
                pw0 = o0 + d0 * FAR_T;  pw1 = o1 + d1 * FAR_T;  pw2 = o2 + d2 * FAR_T;
            } else {
                const float ht = hcl * smem[L_T + (s - 1)];
                const float g0 = (c0 + ht * td0) * a0;
                const float g1 = (c1 + ht * td1) * a1;
                const float g2 = (c2 + ht * td2) * a2;
                pw0 = r00 * g0 + r01 * g1 + r02 * g2 + p0;
                pw1 = r10 * g0 + r11 * g1 + r12 * g2 + p1;
                pw2 = r20 * g0 + r21 * g1 + r22 * g2 + p2;
            }
            // loc = R^T (pw - p)
            const float q0 = pw0 - p0, q1 = pw1 - p1, q2 = pw2 - p2;
            const float l0 = r00 * q0 + r10 * q1 + r20 * q2;
            const float l1 = r01 * q0 + r11 * q1 + r21 * q2;
            const float l2 = r02 * q0 + r12 * q1 + r22 * q2;

            const float X0 = fabsf(l0) * ia0 + EPS_F;
            const float X1 = fabsf(l1) * ia1 + EPS_F;
            const float X2 = fabsf(l2) * ia2 + EPS_F;

            const float A = fpow(X0, exy);
            const float B = fpow(X1, exy);
            const float C = fpow(X2, ez);
            const float f = fpow(A + B, emid) + C;
            const float F = fpow(f, e1);

            // occ = sigmoid(SHARP*(1-F))
            const float z   = SHARP * (1.0f - F);
            const float occ = frcp(1.0f + fexp2(-z * LOG2E));
            cum += occ;
            const float vis = fexp2(cum * (-TAU * LOG2E));

            if (s > 0) {
                const float dx = pw0 - pv0, dy = pw1 - pv1, dz = pw2 - pv2;
                const float dt = fsqrt_(dx * dx + dy * dy + dz * dz);
                depth += 0.5f * (prevvis + vis) * dt;
            }
            pv0 = pw0; pv1 = pw1; pv2 = pw2; prevvis = vis;
        }

        const float dn = mask ? (depth + NEAR_T) : FAR_T;
        best = fminf(best, dn);
    }

    out[pix] = best;
}

// ---------------- launch ----------------
extern "C" void kernel_launch(void* const* d_in, const int* in_sizes, int n_in,
                              void* d_out, int out_size, void* d_ws, size_t ws_size,
                              hipStream_t stream) {
    const float* poses  = (const float*)d_in[0]; // (8,4,4)
    const float* params = (const float*)d_in[1]; // (8,5)
    const float* rays_d = (const float*)d_in[2]; // (360,640,3)
    const float* rays_o = (const float*)d_in[3]; // (3,)
    const float* tvals  = (const float*)d_in[4]; // (10,)
    float* out = (float*)d_out;

    const int threads = 256;
    const int blocks  = (NPIX + threads - 1) / threads;  // 900
    sq_depth_kernel<<<blocks, threads, 0, stream>>>(poses, params, rays_d, rays_o, tvals, out);
}